// SensorAttention_19533511262701
// MI455X (gfx1250) — compile-verified
//
#include <hip/hip_runtime.h>
#include <math.h>

typedef __attribute__((ext_vector_type(2))) float v2f;
typedef __attribute__((ext_vector_type(8))) float v8f;

// D = A(16x4) * B(4x16) + C(16x16), all f32, wave32.
__device__ __forceinline__ v8f wmma_f32(v2f a, v2f b, v8f c) {
  return __builtin_amdgcn_wmma_f32_16x16x4_f32(
      /*neg_a=*/false, a, /*neg_b=*/false, b,
      /*c_mod=*/(short)0, c, /*reuse_a=*/false, /*reuse_b=*/false);
}

// ---------------------------------------------------------------- zero fill
__global__ void zero_kernel(float* __restrict__ p, int n) {
  int i = blockIdx.x * blockDim.x + threadIdx.x;
  if (i < n) p[i] = 0.0f;
}

// ---------------------------------------------------------------- kernel 1
// t[row, c] = tanh( x[row,:] . mlp_w[c,:] + mlp_b[c] ),  row in [0,16384), c in [0,8)
// One wave per row, 8 rows per block; mlp_w (32KB) staged in LDS per block.
__global__ __launch_bounds__(256) void mlp_kernel(const float* __restrict__ x,
                                                  const float* __restrict__ W,
                                                  const float* __restrict__ bias,
                                                  float* __restrict__ t) {
  __shared__ float Wl[8 * 1024];
  int tid = threadIdx.x;
  const float4* W4 = (const float4*)W;
  float4* Wl4 = (float4*)Wl;
  for (int i = tid; i < 2048; i += 256) Wl4[i] = W4[i];
  __syncthreads();

  int wave = tid >> 5, lane = tid & 31;
  int row = blockIdx.x * 8 + wave;                // 2048 blocks * 8 = 16384 rows
  const float4* xr = (const float4*)(x + (size_t)row * 1024);

  float acc[8];
#pragma unroll
  for (int c = 0; c < 8; ++c) acc[c] = 0.0f;

  for (int i = 0; i < 8; ++i) {
    int wi = i * 32 + lane;                       // float4 index within row
    float4 xv = xr[wi];
#pragma unroll
    for (int c = 0; c < 8; ++c) {
      float4 wv = Wl4[c * 256 + wi];
      acc[c] += xv.x * wv.x + xv.y * wv.y + xv.z * wv.z + xv.w * wv.w;
    }
  }
#pragma unroll
  for (int c = 0; c < 8; ++c)
#pragma unroll
    for (int off = 16; off > 0; off >>= 1)
      acc[c] += __shfl_xor(acc[c], off, 32);

  if (lane == 0) {
#pragma unroll
    for (int c = 0; c < 8; ++c)
      t[(size_t)row * 8 + c] = tanhf(acc[c] + bias[c]);
  }
}

// ---------------------------------------------------------------- kernel 2
// T[bh, e, g] += sum_l Q[b,l,h,e] * relu(a[b,h,l] * selW[h,g])   (split-K x4)
__global__ __launch_bounds__(128) void tmat_kernel(const float* __restrict__ Q,
                                                   const float* __restrict__ selW,
                                                   const float* __restrict__ t,
                                                   float* __restrict__ T) {
  int wg = blockIdx.x;                   // B*H*4 = 512
  int part = wg & 3;
  int bh = wg >> 2;
  int b = bh >> 3, h = bh & 7;

  __shared__ float a[256];
  int tid = threadIdx.x;
  const float* tp = t + (size_t)b * 8192 + (size_t)h * 1024 + part * 256;
  for (int i = tid; i < 256; i += 128) a[i] = tp[i];
  __syncthreads();

  int wave = tid >> 5, lane = tid & 31;
  int half = lane >> 4, lm = lane & 15;
  int e = wave * 16 + lm;                          // A-matrix M index (e)

  float wsel[4];
#pragma unroll
  for (int gi = 0; gi < 4; ++gi) wsel[gi] = selW[h * 64 + gi * 16 + lm];

  v8f acc[4] = {};
  // Q element: (b*1024 + l)*512 + h*64 + e ; l = part*256 + ll
  const float* Qbase = Q + ((size_t)b * 1024 + (size_t)part * 256) * 512 + h * 64 + e;

  for (int ll = 0; ll < 256; ll += 4) {
    int la = ll + half * 2;
    v2f af;
    af.x = Qbase[(size_t)la * 512];
    af.y = Qbase[(size_t)(la + 1) * 512];
    float a0 = a[la], a1 = a[la + 1];
#pragma unroll
    for (int gi = 0; gi < 4; ++gi) {
      float s0 = a0 * wsel[gi], s1 = a1 * wsel[gi];
      v2f bf;
      bf.x = s0 > 0.0f ? s0 : 0.0f;
      bf.y = s1 > 0.0f ? s1 : 0.0f;
      acc[gi] = wmma_f32(af, bf, acc[gi]);
    }
  }

  float* Tp = T + (size_t)bh * 4096;
#pragma unroll
  for (int gi = 0; gi < 4; ++gi)
#pragma unroll
    for (int jj = 0; jj < 8; ++jj) {
      int er = wave * 16 + jj + half * 8;
      int g = gi * 16 + lm;
      atomicAdd(&Tp[er * 64 + g], acc[gi][jj]);
    }
}

// ---------------------------------------------------------------- kernel 3
// Per (b,h, s-quarter): scores2 = K_tile @ T ; series = softmax(0.125*scores2, g) ;
// V[d,g] += values_tile^T @ series ; atomically accumulated into out[b,d,h,g].
#define TPAD 72
#define KPAD 65
__global__ __launch_bounds__(128) void attn_kernel(const float* __restrict__ Km,
                                                   const float* __restrict__ Vm,
                                                   const float* __restrict__ Tg,
                                                   float* __restrict__ out) {
  int wg = blockIdx.x;                   // B*H*4 = 512
  int part = wg & 3;
  int bh = wg >> 2;
  int b = bh >> 3, h = bh & 7;

  __shared__ float Tl[64 * TPAD];        // 18.0 KB
  __shared__ float Kb[64 * KPAD];        // 16.25 KB
  __shared__ float Sb[64 * KPAD];        // 16.25 KB

  int tid = threadIdx.x;
  const float* Tp = Tg + (size_t)bh * 4096;
  for (int i = tid; i < 4096; i += 128) {
    int e = i >> 6, g = i & 63;
    Tl[e * TPAD + g] = Tp[i];
  }

  int wave = tid >> 5, lane = tid & 31;
  int half = lane >> 4, lm = lane & 15;

  v8f vacc[4] = {};
  const float* Kbase = Vm == Km ? Km : Km;  // keep compiler honest (no-op)
  Kbase = Km + ((size_t)b * 1024 * 8 + h) * 64;          // + l*512 + e
  const float* Vbase = Vm + ((size_t)b * 1024 * 8 + h) * 64;  // + l*512 + d

  for (int chunk = 0; chunk < 4; ++chunk) {
    int s0 = part * 256 + chunk * 64;
    __syncthreads();   // previous chunk fully consumed Kb/Sb; Tl ready (chunk 0)

    // stage K tile [64 rows x 64 e] into padded LDS (coalesced)
    for (int i = tid; i < 4096; i += 128) {
      int r = i >> 6, e = i & 63;
      Kb[r * KPAD + e] = Kbase[(size_t)(s0 + r) * 512 + e];
    }
    __syncthreads();

    // phase 1: scores rows [wave*16, wave*16+16) x all 64 g, contract over e
    v8f sacc[4] = {};
    int m = wave * 16 + lm;
    for (int e0 = 0; e0 < 64; e0 += 4) {
      int ka = e0 + half * 2;
      v2f af;
      af.x = Kb[m * KPAD + ka];
      af.y = Kb[m * KPAD + ka + 1];
#pragma unroll
      for (int gi = 0; gi < 4; ++gi) {
        int n = gi * 16 + lm;
        v2f bf;
        bf.x = Tl[ka * TPAD + n];
        bf.y = Tl[(ka + 1) * TPAD + n];
        sacc[gi] = wmma_f32(af, bf, sacc[gi]);
      }
    }
#pragma unroll
    for (int gi = 0; gi < 4; ++gi)
#pragma unroll
      for (int jj = 0; jj < 8; ++jj) {
        int r = wave * 16 + jj + half * 8;
        Sb[r * KPAD + gi * 16 + lm] = sacc[gi][jj];
      }
    __syncthreads();

    // softmax over g (scale = 1/sqrt(64) = 0.125), one thread per row
    if (tid < 64) {
      float* row = &Sb[tid * KPAD];
      float mx = row[0] * 0.125f;
      for (int g = 1; g < 64; ++g) mx = fmaxf(mx, row[g] * 0.125f);
      float sum = 0.0f;
      for (int g = 0; g < 64; ++g) {
        float ev = __expf(row[g] * 0.125f - mx);
        row[g] = ev;
        sum += ev;
      }
      float inv = 1.0f / sum;
      for (int g = 0; g < 64; ++g) row[g] *= inv;
    }
    __syncthreads();

    // phase 2: V[d,g] += sum over 64 l of values[b,l,h,d]*series[l,g]
    const float* Vp = Vbase + (size_t)s0 * 512 + wave * 16 + lm;
    for (int k0 = 0; k0 < 64; k0 += 4) {
      int ka = k0 + half * 2;
      v2f af;
      af.x = Vp[(size_t)ka * 512];
      af.y = Vp[(size_t)(ka + 1) * 512];
#pragma unroll
      for (int gi = 0; gi < 4; ++gi) {
        int n = gi * 16 + lm;
        v2f bf;
        bf.x = Sb[ka * KPAD + n];
        bf.y = Sb[(ka + 1) * KPAD + n];
        vacc[gi] = wmma_f32(af, bf, vacc[gi]);
      }
    }
  }

  // accumulate partial V into out[b, d, h, g]  (split over 4 s-parts -> atomics)
#pragma unroll
  for (int gi = 0; gi < 4; ++gi)
#pragma unroll
    for (int jj = 0; jj < 8; ++jj) {
      int d = wave * 16 + jj + half * 8;
      int g = gi * 16 + lm;
      atomicAdd(&out[(((size_t)b * 64 + d) * 8 + h) * 64 + g], vacc[gi][jj]);
    }
}

// ---------------------------------------------------------------- launcher
extern "C" void kernel_launch(void* const* d_in, const int* in_sizes, int n_in,
                              void* d_out, int out_size, void* d_ws, size_t ws_size,
                              hipStream_t stream) {
  const float* queries = (const float*)d_in[0];  // [16,1024,8,64]
  const float* keys    = (const float*)d_in[1];  // [16,1024,8,64]
  const float* values  = (const float*)d_in[2];  // [16,1024,8,64]
  const float* x       = (const float*)d_in[3];  // [16,1024,1024]
  const float* mlp_w   = (const float*)d_in[4];  // [8,1024]
  const float* mlp_b   = (const float*)d_in[5];  // [8]
  const float* sel_W   = (const float*)d_in[6];  // [8,64,1]
  (void)d_in; (void)in_sizes; (void)n_in; (void)ws_size;

  float* out  = (float*)d_out;                   // [16,64,8,64] = 524288
  float* t_ws = (float*)d_ws;                    // 131072 floats
  float* T_ws = t_ws + 131072;                   // 128*64*64 = 524288 floats

  // zero the atomic accumulation targets
  zero_kernel<<<2048, 256, 0, stream>>>(out, 524288);
  zero_kernel<<<2048, 256, 0, stream>>>(T_ws, 524288);

  // t = tanh(x @ mlp_w^T + mlp_b)
  mlp_kernel<<<2048, 256, 0, stream>>>(x, mlp_w, mlp_b, t_ws);

  // T[b,h] = Q^T @ relu(a * selW)   (split-K over l, 4 parts)
  tmat_kernel<<<512, 128, 0, stream>>>(queries, sel_W, t_ws, T_ws);

  // scores2 = K @ T ; softmax ; V += values^T @ series
  attn_kernel<<<512, 128, 0, stream>>>(keys, values, T_ws, out);
}